// Squad_Task_50508815401456
// MI455X (gfx1250) — compile-verified
//
#include <hip/hip_runtime.h>
#include <hip/hip_bf16.h>
#include <math.h>

typedef __attribute__((ext_vector_type(2))) float v2f;
typedef __attribute__((ext_vector_type(8))) float v8f;

#define BB  64
#define SS  512
#define DD  1024

// ---------------------------------------------------------------------------
// Phase 1: logits via V_WMMA_F32_16X16X4_F32.
// One wave32 owns 16 rows of hidden. A = 16x4 fp32 tile of hidden,
// B = 4x16 with col0 = w_start chunk, col1 = w_end chunk, cols 2..15 zero.
// Accumulate over K=1024 (256 WMMAs, unrolled x8). Lanes {0,1,16,17} end up
// holding the 16 start/end dot products in their 8 accumulator VGPRs.
// ---------------------------------------------------------------------------
__global__ __launch_bounds__(256)
void logits_wmma_kernel(const float* __restrict__ hidden,
                        const float* __restrict__ w_start,
                        const float* __restrict__ w_end,
                        const int*   __restrict__ labels,
                        float* __restrict__ s_logits,
                        float* __restrict__ e_logits)
{
    const int lane = threadIdx.x & 31;
    const int wave = threadIdx.x >> 5;
    const int n    = lane & 15;     // N / M-in-halfwave index
    const int half = lane >> 4;     // 0: K=0,1  1: K=2,3 (A and B mirror layout)
    const int row0 = blockIdx.x * 128 + wave * 16;
    const int row  = row0 + n;      // this lane's A-matrix row

    const float* aptr = hidden + (size_t)row * DD + 2 * half;
    const float* wptr = ((n == 1) ? w_end : w_start) + 2 * half;
    const float  colmask = (n < 2) ? 1.0f : 0.0f;   // zero B columns 2..15

    v8f acc = {};
    for (int k0 = 0; k0 < DD; k0 += 32) {
        v2f av[8], bv[8];
#pragma unroll
        for (int u = 0; u < 8; ++u) {
            av[u] = *(const v2f*)(aptr + k0 + 4 * u);     // {A[row,k+2h], A[row,k+2h+1]}
            v2f w = *(const v2f*)(wptr + k0 + 4 * u);     // {B[2h,n],     B[2h+1,n]}
            bv[u].x = w.x * colmask;
            bv[u].y = w.y * colmask;
        }
#pragma unroll
        for (int u = 0; u < 8; ++u) {
            acc = __builtin_amdgcn_wmma_f32_16x16x4_f32(
                false, av[u], false, bv[u], (short)0, acc, false, false);
        }
    }

    // C/D layout: VGPR r, lane(n,half) = D[M = r + 8*half, N = n].
    // Column 0 = start dots, column 1 = end dots; scale by label scalar.
    const int batch = row0 / SS;   // 16 rows never straddle a batch (512 % 16 == 0)
    if (n < 2) {
        const float scale = (float)labels[batch * 2 + n];
        float* outp = (n == 0) ? s_logits : e_logits;
        const int base = row0 + 8 * half;
#pragma unroll
        for (int r = 0; r < 8; ++r)
            outp[base + r] = scale * acc[r];
    }
}

// ---------------------------------------------------------------------------
// Phase 2: two softmaxes over S=512 per batch. One block per batch.
// ---------------------------------------------------------------------------
__global__ __launch_bounds__(512)
void softmax_kernel(const float* __restrict__ s_logits,
                    const float* __restrict__ e_logits,
                    float* __restrict__ p_start,
                    float* __restrict__ p_end)
{
    __shared__ float red[SS];
    const int b = blockIdx.x;
    const int t = threadIdx.x;

    const float xs = s_logits[b * SS + t];
    const float xe = e_logits[b * SS + t];

    // start softmax
    red[t] = xs; __syncthreads();
    for (int off = 256; off > 0; off >>= 1) {
        if (t < off) red[t] = fmaxf(red[t], red[t + off]);
        __syncthreads();
    }
    float mx = red[0]; __syncthreads();
    float es = expf(xs - mx);
    red[t] = es; __syncthreads();
    for (int off = 256; off > 0; off >>= 1) {
        if (t < off) red[t] += red[t + off];
        __syncthreads();
    }
    float sum = red[0]; __syncthreads();
    p_start[b * SS + t] = es / sum;

    // end softmax
    red[t] = xe; __syncthreads();
    for (int off = 256; off > 0; off >>= 1) {
        if (t < off) red[t] = fmaxf(red[t], red[t + off]);
        __syncthreads();
    }
    mx = red[0]; __syncthreads();
    float ee = expf(xe - mx);
    red[t] = ee; __syncthreads();
    for (int off = 256; off > 0; off >>= 1) {
        if (t < off) red[t] += red[t + off];
        __syncthreads();
    }
    sum = red[0];
    p_end[b * SS + t] = ee / sum;
}

// ---------------------------------------------------------------------------
// Phase 3: materialize pred[b,r,c] = p_start[c] + p_end[r], lower triangle,
// with float4 (b128) stores — this is the 67 MB bandwidth-bound pass.
// One block per (b, r) row; 128 threads x 4 columns.
// ---------------------------------------------------------------------------
__global__ __launch_bounds__(128)
void pred_kernel(const float* __restrict__ p_start,
                 const float* __restrict__ p_end,
                 float* __restrict__ pred)
{
    const int r  = blockIdx.x & (SS - 1);
    const int b  = blockIdx.x >> 9;
    const int c0 = threadIdx.x * 4;

    const float per = p_end[b * SS + r];
    const float4 ps = *(const float4*)(p_start + b * SS + c0);

    float4 o;
    o.x = (c0 + 0 <= r) ? ps.x + per : 0.0f;
    o.y = (c0 + 1 <= r) ? ps.y + per : 0.0f;
    o.z = (c0 + 2 <= r) ? ps.z + per : 0.0f;
    o.w = (c0 + 3 <= r) ? ps.w + per : 0.0f;

    *(float4*)(pred + ((size_t)b * SS + r) * SS + c0) = o;
}

// ---------------------------------------------------------------------------
// Phase 4: argmax over the lower triangle without touching the SxS matrix:
// prefix-argmax of p_start fused with a scan over r. Strict '>' updates
// reproduce jnp.argmax first-occurrence (lowest flat index r*S+c) semantics.
// ---------------------------------------------------------------------------
__global__ __launch_bounds__(64)
void argmax_kernel(const float* __restrict__ p_start,
                   const float* __restrict__ p_end,
                   float* __restrict__ final_pred)
{
    const int b = threadIdx.x;
    if (b >= BB) return;
    const float* ps = p_start + b * SS;
    const float* pe = p_end + b * SS;

    float best_ps = ps[0];            // running prefix max of p_start
    int   best_c  = 0;
    float bestv   = pe[0] + best_ps;  // r = 0 candidate
    int   br = 0, bc = 0;

    for (int r = 1; r < SS; ++r) {
        const float x = ps[r];
        if (x > best_ps) { best_ps = x; best_c = r; }   // earliest c on ties
        const float v = pe[r] + best_ps;
        if (v > bestv) { bestv = v; br = r; bc = best_c; }  // earliest r on ties
    }
    final_pred[b * 2 + 0] = (float)br;  // end_idx  = argmax // S
    final_pred[b * 2 + 1] = (float)bc;  // start_idx = argmax %  S
}

// ---------------------------------------------------------------------------
extern "C" void kernel_launch(void* const* d_in, const int* in_sizes, int n_in,
                              void* d_out, int out_size, void* d_ws, size_t ws_size,
                              hipStream_t stream)
{
    (void)in_sizes; (void)n_in; (void)out_size; (void)ws_size;

    const float* hidden  = (const float*)d_in[0];
    const float* w_start = (const float*)d_in[1];
    const float* w_end   = (const float*)d_in[2];
    const int*   labels  = (const int*)d_in[3];

    float* pred   = (float*)d_out;                       // [B, S, S]
    float* finalp = pred + (size_t)BB * SS * SS;         // [B, 2]

    float* ws       = (float*)d_ws;                      // 4 * 32768 floats = 512 KB
    float* s_logits = ws + 0 * BB * SS;
    float* e_logits = ws + 1 * BB * SS;
    float* p_start  = ws + 2 * BB * SS;
    float* p_end    = ws + 3 * BB * SS;

    // 32768 rows / 16 rows-per-wave / 8 waves-per-block = 256 blocks
    logits_wmma_kernel<<<(BB * SS) / 128, 256, 0, stream>>>(
        hidden, w_start, w_end, labels, s_logits, e_logits);

    softmax_kernel<<<BB, SS, 0, stream>>>(s_logits, e_logits, p_start, p_end);

    pred_kernel<<<BB * SS, 128, 0, stream>>>(p_start, p_end, pred);

    argmax_kernel<<<1, 64, 0, stream>>>(p_start, p_end, finalp);
}